// HypersphericalGraphEncoder_55138790146370
// MI455X (gfx1250) — compile-verified
//
#include <hip/hip_runtime.h>
#include <hip/hip_bf16.h>

// ---------------------------------------------------------------------------
// HypersphericalGraphEncoder for MI455X (gfx1250, wave32).
// GEMMs use V_WMMA_F32_16X16X4_F32 (fp32-exact matrix path).
// Scatter/aggregate uses global_atomic_add_f32; agg buffers are L2-resident.
// ---------------------------------------------------------------------------

typedef float v2f __attribute__((ext_vector_type(2)));
typedef float v8f __attribute__((ext_vector_type(8)));

#define IN_DIM 128
#define HID 256
#define LAT 64

// ---------------- degree / norm ----------------

__global__ void k_deg_init(float* __restrict__ deg, int n) {
  int i = blockIdx.x * blockDim.x + threadIdx.x;
  if (i < n) deg[i] = 1.0f;  // self loop
}

__global__ void k_deg_count(const int* __restrict__ ei, float* __restrict__ deg, int E) {
  int e = blockIdx.x * blockDim.x + threadIdx.x;
  if (e < E) atomicAdd(&deg[ei[E + e]], 1.0f);  // dst = edge_index[1]
}

__global__ void k_deg_to_dis(float* __restrict__ deg, int n) {
  int i = blockIdx.x * blockDim.x + threadIdx.x;
  if (i < n) deg[i] = rsqrtf(deg[i]);  // deg >= 1, exact per reference deg^-0.5
}

// ---------------- WMMA fp32 GEMM: C[M,Nc] = A[M,K] @ B[K,Nc] ----------------
// One wave per 16x16 output tile. M must be a multiple of 16 (50000 = 3125*16).
// A-layout (16x4 f32): lanes 0-15 -> rows, half 0 holds K=k0,k0+1; half 1 K=k0+2,k0+3.
// B-layout (4x16 f32): lanes 0-15 -> cols, VGPR r / half h holds K = k0 + r + 2h.
// C/D (16x16 f32): VGPR r, half h -> row r + 8h, col = lane&15.

__global__ void k_gemm_wmma_f32(const float* __restrict__ A,
                                const float* __restrict__ B,
                                float* __restrict__ C,
                                int M, int K, int Nc) {
  const int wavesPerBlock = blockDim.x >> 5;
  const int wave = blockIdx.x * wavesPerBlock + (threadIdx.x >> 5);
  const int tilesN = Nc >> 4;
  const int tilesM = M >> 4;
  const int tileM = wave / tilesN;
  const int tileN = wave % tilesN;
  if (tileM >= tilesM) return;  // wave-uniform: EXEC stays all-ones for WMMA

  const int lane = threadIdx.x & 31;
  const int half = lane >> 4;   // 0 or 1
  const int l    = lane & 15;

  const float* Arow = A + (long)(tileM * 16 + l) * K + 2 * half;  // K=k0+2h, +1
  const float* Bcol = B + (tileN * 16 + l);

  v8f acc = (v8f){0.f, 0.f, 0.f, 0.f, 0.f, 0.f, 0.f, 0.f};

  for (int k0 = 0; k0 < K; k0 += 4) {
    v2f a = *(const v2f*)(Arow + k0);          // global_load_b64
    v2f b;
    b.x = Bcol[(long)(k0 + 2 * half + 0) * Nc];
    b.y = Bcol[(long)(k0 + 2 * half + 1) * Nc];
    acc = __builtin_amdgcn_wmma_f32_16x16x4_f32(
        /*neg_a=*/false, a, /*neg_b=*/false, b,
        /*c_mod=*/(short)0, acc, /*reuse_a=*/false, /*reuse_b=*/false);
  }

  float* Ctile = C + (long)(tileM * 16) * Nc + tileN * 16 + l;
#pragma unroll
  for (int r = 0; r < 8; ++r) {
    Ctile[(long)(r + 8 * half) * Nc] = acc[r];
  }
}

// ---------------- self-loop seed: agg[i,:] = m[i,:] * dis[i]^2 ----------------

__global__ void k_selfloop_seed(const float* __restrict__ m,
                                const float* __restrict__ dis,
                                float* __restrict__ agg, int n, int Cc) {
  long tid = (long)blockIdx.x * blockDim.x + threadIdx.x;
  int cpr = Cc >> 2;  // float4 chunks per row
  long row = tid / cpr;
  if (row >= n) return;
  int c4 = (int)(tid % cpr) << 2;
  float d = dis[row];
  float nn = d * d;
  float4 v = *(const float4*)(m + row * Cc + c4);
  v.x *= nn; v.y *= nn; v.z *= nn; v.w *= nn;
  *(float4*)(agg + row * Cc + c4) = v;
}

// ---------------- edge scatter: agg[dst,:] += m[src,:] * dis[src]*dis[dst] ----

__global__ void k_scatter_edges(const int* __restrict__ ei,
                                const float* __restrict__ m,
                                const float* __restrict__ dis,
                                float* __restrict__ agg, int E, int Cc) {
  long tid = (long)blockIdx.x * blockDim.x + threadIdx.x;
  int tpe = Cc >> 2;  // threads per edge, float4 each
  long e = tid / tpe;
  if (e >= E) return;
  int c4 = (int)(tid % tpe) << 2;
  int s = ei[e];        // edge_index[0][e] : message source
  int d = ei[E + e];    // edge_index[1][e] : aggregation target
  float norm = dis[s] * dis[d];
  float4 v = *(const float4*)(m + (long)s * Cc + c4);
  float* out = agg + (long)d * Cc + c4;
  atomicAdd(out + 0, v.x * norm);
  atomicAdd(out + 1, v.y * norm);
  atomicAdd(out + 2, v.z * norm);
  atomicAdd(out + 3, v.w * norm);
}

// ---------------- bias + relu (in place) ----------------

__global__ void k_bias_relu(float* __restrict__ agg, const float* __restrict__ b,
                            int n, int Cc) {
  long tid = (long)blockIdx.x * blockDim.x + threadIdx.x;
  int cpr = Cc >> 2;
  long row = tid / cpr;
  if (row >= n) return;
  int c4 = (int)(tid % cpr) << 2;
  float4 bb = *(const float4*)(b + c4);
  float4 v = *(const float4*)(agg + row * Cc + c4);
  v.x = fmaxf(v.x + bb.x, 0.f);
  v.y = fmaxf(v.y + bb.y, 0.f);
  v.z = fmaxf(v.z + bb.z, 0.f);
  v.w = fmaxf(v.w + bb.w, 0.f);
  *(float4*)(agg + row * Cc + c4) = v;
}

// ---------------- bias + row-wise L2 normalize (wave32 per row, 64 cols) -----

__global__ void k_finalize_norm(const float* __restrict__ agg,
                                const float* __restrict__ b,
                                float* __restrict__ out, int n) {
  int wave = blockIdx.x * (blockDim.x >> 5) + (threadIdx.x >> 5);
  if (wave >= n) return;
  int lane = threadIdx.x & 31;
  float2 v = *(const float2*)(agg + (long)wave * LAT + lane * 2);
  float2 bb = *(const float2*)(b + lane * 2);
  v.x += bb.x; v.y += bb.y;
  float ss = v.x * v.x + v.y * v.y;
#pragma unroll
  for (int m = 16; m >= 1; m >>= 1) ss += __shfl_xor(ss, m, 32);
  float inv = 1.0f / (sqrtf(ss) + 1e-8f);
  float2 o; o.x = v.x * inv; o.y = v.y * inv;
  *(float2*)(out + (long)wave * LAT + lane * 2) = o;
}

// ---------------- launch ----------------

static inline int ceil_div(long a, long b) { return (int)((a + b - 1) / b); }

extern "C" void kernel_launch(void* const* d_in, const int* in_sizes, int n_in,
                              void* d_out, int out_size, void* d_ws, size_t ws_size,
                              hipStream_t stream) {
  const float* x  = (const float*)d_in[0];
  const int*   ei = (const int*)d_in[1];
  const float* W1 = (const float*)d_in[2];
  const float* b1 = (const float*)d_in[3];
  const float* W2 = (const float*)d_in[4];
  const float* b2 = (const float*)d_in[5];
  float* out = (float*)d_out;

  const int N = in_sizes[0] / IN_DIM;   // 50000
  const int E = in_sizes[1] / 2;        // 800000

  // Workspace layout (floats). Total = N*(1+256+256+64+64) ~= 128.2 MB.
  float* ws   = (float*)d_ws;
  float* dis  = ws;                       // [N]      degree -> deg^-1/2
  float* m1   = dis  + N;                 // [N,HID]  x @ W1
  float* agg1 = m1   + (size_t)N * HID;   // [N,HID]  aggregated, then relu'd h
  float* m2   = agg1 + (size_t)N * HID;   // [N,LAT]  h @ W2
  float* agg2 = m2   + (size_t)N * LAT;   // [N,LAT]  aggregated

  const int BT = 256;

  // 1) degrees (self loop = init 1) -> dis = deg^-1/2
  k_deg_init<<<ceil_div(N, BT), BT, 0, stream>>>(dis, N);
  k_deg_count<<<ceil_div(E, BT), BT, 0, stream>>>(ei, dis, E);
  k_deg_to_dis<<<ceil_div(N, BT), BT, 0, stream>>>(dis, N);

  // 2) m1 = x @ W1   (tiles: (N/16) * (HID/16) waves, 8 waves per block)
  {
    long waves = (long)(N / 16) * (HID / 16);
    k_gemm_wmma_f32<<<ceil_div(waves, BT / 32), BT, 0, stream>>>(x, W1, m1, N, IN_DIM, HID);
  }

  // 3) agg1 = selfloop + edge scatter; then h = relu(agg1 + b1) in place
  k_selfloop_seed<<<ceil_div((long)N * (HID / 4), BT), BT, 0, stream>>>(m1, dis, agg1, N, HID);
  k_scatter_edges<<<ceil_div((long)E * (HID / 4), BT), BT, 0, stream>>>(ei, m1, dis, agg1, E, HID);
  k_bias_relu<<<ceil_div((long)N * (HID / 4), BT), BT, 0, stream>>>(agg1, b1, N, HID);

  // 4) m2 = h @ W2
  {
    long waves = (long)(N / 16) * (LAT / 16);
    k_gemm_wmma_f32<<<ceil_div(waves, BT / 32), BT, 0, stream>>>(agg1, W2, m2, N, HID, LAT);
  }

  // 5) agg2 = selfloop + edge scatter; then out = normalize(agg2 + b2)
  k_selfloop_seed<<<ceil_div((long)N * (LAT / 4), BT), BT, 0, stream>>>(m2, dis, agg2, N, LAT);
  k_scatter_edges<<<ceil_div((long)E * (LAT / 4), BT), BT, 0, stream>>>(ei, m2, dis, agg2, E, LAT);
  k_finalize_norm<<<ceil_div(N, BT / 32), BT, 0, stream>>>(agg2, b2, out, N);
}